// GCN_NET_14774687498576
// MI455X (gfx1250) — compile-verified
//
#include <hip/hip_runtime.h>

typedef __attribute__((ext_vector_type(2))) float v2f;
typedef __attribute__((ext_vector_type(8))) float v8f;

// ---------------- utility kernels ----------------
__global__ void fill_f32(float* __restrict__ p, float v, long long n) {
    long long i = (long long)blockIdx.x * blockDim.x + threadIdx.x;
    if (i < n) p[i] = v;
}

__global__ void deg_accum(const int* __restrict__ dst, float* __restrict__ deg, int E) {
    int e = blockIdx.x * blockDim.x + threadIdx.x;
    if (e < E) atomicAdd(&deg[dst[e]], 1.0f);
}

__global__ void deg_to_dinv(float* __restrict__ deg, int n) {
    int i = blockIdx.x * blockDim.x + threadIdx.x;
    if (i < n) {
        float d = deg[i];
        deg[i] = d > 0.0f ? rsqrtf(d) : 0.0f;
    }
}

// Pack W[wrows,wcols] -> Wt[64][kdim], transposed + zero padded, so the GEMM
// inner loop is branch-free and B fragments are contiguous 8-byte LDS loads.
__global__ void pack_wt(const float* __restrict__ W, int wrows, int wcols,
                        float* __restrict__ Wt, int kdim) {
    int i = blockIdx.x * blockDim.x + threadIdx.x;   // i = col*kdim + k
    if (i >= 64 * kdim) return;
    int col = i / kdim;
    int k   = i - col * kdim;
    Wt[i] = (k < wrows && col < wcols) ? W[(long long)k * wcols + col] : 0.0f;
}

// ---------------- fp32 WMMA GEMM ----------------
// Out[nrows,64] = A[nrows,lda] (first KDIM cols) @ Wpad[KDIM,64]
// Wt is the transposed padded weight [64][KDIM]; staged once into LDS per block.
// One wave per 16-row tile; 4 column tiles (64 cols) share each A fragment.
template <int KDIM>
__global__ void wmma_gemm_f32(const float* __restrict__ A, int lda,
                              const float* __restrict__ Wt,
                              float* __restrict__ Out, int nrows) {
    __shared__ __align__(16) float wlds[64 * KDIM];
    for (int i = threadIdx.x; i < 64 * KDIM; i += blockDim.x)
        wlds[i] = Wt[i];
    __syncthreads();

    const int lane = threadIdx.x & 31;
    const int half = lane >> 4;   // 0: lanes 0-15, 1: lanes 16-31
    const int lp   = lane & 15;
    const int wave = blockIdx.x * (blockDim.x >> 5) + (threadIdx.x >> 5);
    const int m0   = wave * 16;
    if (m0 >= nrows) return;      // wave-uniform: EXEC stays all-1s for WMMA

    int arow = m0 + lp;
    if (arow >= nrows) arow = nrows - 1;              // clamp; stores guarded below
    const v2f* __restrict__ ap = (const v2f*)(A + (long long)arow * lda);

    // B fragment sources: Wt[col][kb], Wt[col][kb+1] contiguous -> one b64 LDS load.
    const v2f* __restrict__ w0 = (const v2f*)&wlds[(lp +  0) * KDIM];
    const v2f* __restrict__ w1 = (const v2f*)&wlds[(lp + 16) * KDIM];
    const v2f* __restrict__ w2 = (const v2f*)&wlds[(lp + 32) * KDIM];
    const v2f* __restrict__ w3 = (const v2f*)&wlds[(lp + 48) * KDIM];

    v8f c0 = {}, c1 = {}, c2 = {}, c3 = {};

#pragma unroll 4
    for (int k = 0; k < KDIM; k += 4) {
        const int idx = (k >> 1) + half;              // (k + 2*half) / 2, kb even
        const v2f a = ap[idx];                        // A: lanes16-31 hold K+2,K+3
        c0 = __builtin_amdgcn_wmma_f32_16x16x4_f32(false, a, false, w0[idx], (short)0, c0, false, false);
        c1 = __builtin_amdgcn_wmma_f32_16x16x4_f32(false, a, false, w1[idx], (short)0, c1, false, false);
        c2 = __builtin_amdgcn_wmma_f32_16x16x4_f32(false, a, false, w2[idx], (short)0, c2, false, false);
        c3 = __builtin_amdgcn_wmma_f32_16x16x4_f32(false, a, false, w3[idx], (short)0, c3, false, false);
    }

    // C/D layout: VGPR v -> M = v + 8*half, N = lp (+16 per tile)
    const v8f acc[4] = {c0, c1, c2, c3};
    float* __restrict__ orow = Out + (long long)(m0 + 8 * half) * 64 + lp;

    if (m0 + 16 <= nrows) {
        // Complete tile: wave-uniform scalar branch, no per-store EXEC masking.
#pragma unroll
        for (int t = 0; t < 4; ++t)
#pragma unroll
            for (int v = 0; v < 8; ++v)
                orow[(long long)v * 64 + t * 16] = acc[t][v];
    } else {
#pragma unroll
        for (int t = 0; t < 4; ++t)
#pragma unroll
            for (int v = 0; v < 8; ++v) {
                const int row = m0 + v + 8 * half;
                if (row < nrows) Out[(long long)row * 64 + t * 16 + lp] = acc[t][v];
            }
    }
}

// ---------------- edge scatter (message passing) ----------------
// 64 threads per edge: coalesced 256B gather of H[src], L2-resident f32 atomics into Acc[dst].
__global__ void scatter_edges(const int* __restrict__ src, const int* __restrict__ dst,
                              const float* __restrict__ dinv,
                              const float* __restrict__ H,      // stride 64
                              float* __restrict__ Acc, int accStride,
                              int E, int feat) {
    long long tid = (long long)blockIdx.x * blockDim.x + threadIdx.x;
    const int f = (int)(tid & 63);
    const long long e = tid >> 6;
    if (e >= E || f >= feat) return;
    const int s = src[e];
    const int d = dst[e];
    const float nrm = dinv[s] * dinv[d];
    atomicAdd(&Acc[(long long)d * accStride + f], H[(long long)s * 64 + f] * nrm);
}

// ---------------- layer-1 finalize: self-loop + bias + ReLU, keep 64-padding zeroed ----
__global__ void finalize_relu(float* __restrict__ Acc, const float* __restrict__ H,
                              const float* __restrict__ dinv, const float* __restrict__ bias,
                              int n, int feat) {
    long long tid = (long long)blockIdx.x * blockDim.x + threadIdx.x;
    const int f = (int)(tid & 63);
    const long long i = tid >> 6;
    if (i >= n) return;
    float v = 0.0f;
    if (f < feat) {
        const float di = dinv[i];
        v = Acc[i * 64 + f] + H[i * 64 + f] * di * di + bias[f];
        v = fmaxf(v, 0.0f);
    }
    Acc[i * 64 + f] = v;
}

// ---------------- layer-2 finalize: self-loop + bias into stride-50 output ----------
__global__ void finalize_out(float* __restrict__ outp, const float* __restrict__ H,
                             const float* __restrict__ dinv, const float* __restrict__ bias,
                             int n, int feat) {
    long long tid = (long long)blockIdx.x * blockDim.x + threadIdx.x;
    const int f = (int)(tid & 63);
    const long long i = tid >> 6;
    if (i >= n || f >= feat) return;
    const float di = dinv[i];
    outp[i * (long long)feat + f] += H[i * 64 + f] * di * di + bias[f];
}

extern "C" void kernel_launch(void* const* d_in, const int* in_sizes, int n_in,
                              void* d_out, int out_size, void* d_ws, size_t ws_size,
                              hipStream_t stream) {
    const float* x   = (const float*)d_in[0];
    const float* W1  = (const float*)d_in[1];
    const float* b1  = (const float*)d_in[2];
    const float* W2  = (const float*)d_in[3];
    const float* b2  = (const float*)d_in[4];
    const int*   ei  = (const int*)d_in[5];   // int32 edge_index [2, E]

    const int FIN = 128, NH = 50, NO = 50, PAD = 64;
    const int n = in_sizes[0] / FIN;          // 100000
    const int E = in_sizes[5] / 2;            // 1600000
    const int* src = ei;
    const int* dst = ei + E;

    // workspace layout
    float* dinv = (float*)d_ws;                              // n floats (deg -> dinv)
    float* wt   = dinv + (((long long)n + 63) & ~63LL);      // 64*128 floats (packed W)
    float* hbuf = wt + 64 * 128;                             // n*64 (GEMM output, both layers)
    float* abuf = hbuf + (long long)n * PAD;                 // n*64 (aggregation accumulator)

    const int B = 256;
    const long long nPad = (long long)n * PAD;
    const long long eThreads = (long long)E * PAD;
    const int gemmBlocks = (int)(((n + 15) / 16 + 7) / 8);   // 8 waves / block, 1 tile / wave

    // ---- degrees (self-loop = init 1.0) ----
    fill_f32<<<(int)((n + B - 1) / B), B, 0, stream>>>(dinv, 1.0f, n);
    deg_accum<<<(int)((E + B - 1) / B), B, 0, stream>>>(dst, dinv, E);
    deg_to_dinv<<<(int)((n + B - 1) / B), B, 0, stream>>>(dinv, n);

    // ---- layer 1: h = x @ W1 (fp32 WMMA) ----
    pack_wt<<<(64 * FIN + B - 1) / B, B, 0, stream>>>(W1, FIN, NH, wt, FIN);
    wmma_gemm_f32<128><<<gemmBlocks, B, 0, stream>>>(x, FIN, wt, hbuf, n);

    // ---- layer 1 aggregation ----
    fill_f32<<<(int)((nPad + B - 1) / B), B, 0, stream>>>(abuf, 0.0f, nPad);
    scatter_edges<<<(int)((eThreads + B - 1) / B), B, 0, stream>>>(src, dst, dinv, hbuf,
                                                                   abuf, PAD, E, NH);
    finalize_relu<<<(int)((nPad + B - 1) / B), B, 0, stream>>>(abuf, hbuf, dinv, b1, n, NH);

    // ---- layer 2: h2 = a1 @ W2 (K padded to 64; abuf cols 50..63 are zero) ----
    pack_wt<<<(64 * PAD + B - 1) / B, B, 0, stream>>>(W2, NH, NO, wt, PAD);
    wmma_gemm_f32<64><<<gemmBlocks, B, 0, stream>>>(abuf, PAD, wt, hbuf, n);

    // ---- layer 2 aggregation into d_out (stride 50) ----
    float* outp = (float*)d_out;
    const long long outN = (long long)n * NO;
    fill_f32<<<(int)((outN + B - 1) / B), B, 0, stream>>>(outp, 0.0f, outN);
    scatter_edges<<<(int)((eThreads + B - 1) / B), B, 0, stream>>>(src, dst, dinv, hbuf,
                                                                   outp, NO, E, NO);
    finalize_out<<<(int)((nPad + B - 1) / B), B, 0, stream>>>(outp, hbuf, dinv, b2, n, NO);
}